// GlobalModel_20203526160534
// MI455X (gfx1250) — compile-verified
//
#include <hip/hip_runtime.h>
#include <hip/hip_bf16.h>

typedef __attribute__((ext_vector_type(2))) float v2f;
typedef __attribute__((ext_vector_type(8))) float v8f;

#define HID 128

// ---------------------------------------------------------------------------
// Node scatter: batch is sorted, so accumulate in registers per graph-run and
// flush one atomic per (graph transition x feature).
// blockDim = 128 (thread t = feature t)
// ---------------------------------------------------------------------------
__global__ void node_scatter_kernel(const float* __restrict__ x,
                                    const int* __restrict__ batch,
                                    float* __restrict__ nsum,
                                    float* __restrict__ ncnt,
                                    int n, int chunk) {
    int t = threadIdx.x;
    long start = (long)blockIdx.x * chunk;
    long end = start + chunk;
    if (end > n) end = n;
    if (start >= end) return;

    int curg = batch[start];
    float acc = 0.0f;
    float cnt = 0.0f;
    for (long i = start; i < end; ++i) {
        int g = batch[i];
        if (g != curg) {
            atomicAdd(&nsum[(size_t)curg * HID + t], acc);
            if (t == 0) atomicAdd(&ncnt[curg], cnt);
            acc = 0.0f; cnt = 0.0f; curg = g;
        }
        acc += x[i * HID + t];
        cnt += 1.0f;
    }
    atomicAdd(&nsum[(size_t)curg * HID + t], acc);
    if (t == 0) atomicAdd(&ncnt[curg], cnt);
}

// ---------------------------------------------------------------------------
// Edge scatter: destinations are random -> per-element atomics (accumulator is
// L2-resident: 8MB << 192MB L2). blockDim = 128 (thread t = feature t).
// ---------------------------------------------------------------------------
__global__ void edge_scatter_kernel(const float* __restrict__ ea,
                                    const int* __restrict__ ei,   // [2,E] flat
                                    const int* __restrict__ batch,
                                    float* __restrict__ esum,
                                    float* __restrict__ ecnt,
                                    int e_total) {
    int t = threadIdx.x;
    for (long e = blockIdx.x; e < e_total; e += gridDim.x) {
        int dst = ei[(long)e_total + e];          // edge_index[1][e]
        int g = batch[dst];
        atomicAdd(&esum[(size_t)g * HID + t], ea[e * HID + t]);
        if (t == 0) atomicAdd(&ecnt[g], 1.0f);
    }
}

// ---------------------------------------------------------------------------
// sums -> means in place:  s / max(c, 1)
// ---------------------------------------------------------------------------
__global__ void finalize_mean_kernel(float* __restrict__ nsum,
                                     float* __restrict__ esum,
                                     const float* __restrict__ ncnt,
                                     const float* __restrict__ ecnt,
                                     long total) {
    long i = (long)blockIdx.x * blockDim.x + threadIdx.x;
    if (i >= total) return;
    long g = i >> 7;  // /128
    nsum[i] = nsum[i] / fmaxf(ncnt[g], 1.0f);
    esum[i] = esum[i] / fmaxf(ecnt[g], 1.0f);
}

// ---------------------------------------------------------------------------
// Fused MLP: relu(concat(u, nmean, emean) @ W1 + b1) @ W2 + b2
// Exact fp32 path via V_WMMA_F32_16X16X4_F32.
// blockDim = 128 (4 waves). Each wave owns a 16-row strip; block covers 64 rows.
//
// 32-bit WMMA fragment layouts (ISA 7.12.2):
//   A 16x4 : lane<16 -> (A[m][k0], A[m][k0+1]); lane>=16 -> (A[m][k0+2], A[m][k0+3])
//   B 4x16 : lane<16 -> (B[k0][n], B[k0+1][n]); lane>=16 -> (B[k0+2][n], B[k0+3][n])
//   C/D    : vgpr i, lane<16 -> D[i][n] ; lane>=16 -> D[8+i][n]
// ---------------------------------------------------------------------------
__global__ void mlp_wmma_kernel(const float* __restrict__ u,
                                const float* __restrict__ nmean,
                                const float* __restrict__ emean,
                                const float* __restrict__ W1,  // [384,128]
                                const float* __restrict__ b1,  // [128]
                                const float* __restrict__ W2,  // [128,128]
                                const float* __restrict__ b2,  // [128]
                                float* __restrict__ out,       // [B,128]
                                int brows) {
    __shared__ float hlds[4 * 16 * HID];   // 32 KB: h strip per wave

    const int lane = threadIdx.x & 31;
    const int wave = threadIdx.x >> 5;
    const int half = lane >> 4;            // 0: K0/K1 rows, 1: K2/K3 rows
    const int l    = lane & 15;            // M index (A) / N index (B)

    const int rowBase = blockIdx.x * 64 + wave * 16;
    if (rowBase >= brows) return;          // wave-uniform: EXEC stays all-ones

    const float* srcs[3] = { u, nmean, emean };

    v8f acc[8];
    v8f zero = {0.f, 0.f, 0.f, 0.f, 0.f, 0.f, 0.f, 0.f};
#pragma unroll
    for (int t = 0; t < 8; ++t) acc[t] = zero;

    const long arow = (long)(rowBase + l) * HID;

    // ---- GEMM1: [16,384] x [384,128], K split into 3 sources of 128 ----
    for (int s = 0; s < 3; ++s) {
        const float* A   = srcs[s];
        const float* W1s = W1 + (size_t)s * HID * HID;   // rows s*128..+127
        for (int kk = 0; kk < 32; ++kk) {
            const int k0 = kk * 4 + 2 * half;
            v2f a;
            a.x = A[arow + k0];
            a.y = A[arow + k0 + 1];
            const float* wr = W1s + (size_t)k0 * HID;
#pragma unroll
            for (int t = 0; t < 8; ++t) {
                v2f b;
                b.x = wr[t * 16 + l];
                b.y = wr[HID + t * 16 + l];
                acc[t] = __builtin_amdgcn_wmma_f32_16x16x4_f32(
                    false, a, false, b, (short)0, acc[t], false, false);
            }
        }
    }

    // ---- bias + relu, stage h strip (16x128) to LDS ----
    float* hb = hlds + wave * 16 * HID;
#pragma unroll
    for (int t = 0; t < 8; ++t) {
        const int col = t * 16 + l;
        const float bias = b1[col];
#pragma unroll
        for (int i = 0; i < 8; ++i) {
            float v = acc[t][i] + bias;
            hb[(i + 8 * half) * HID + col] = fmaxf(v, 0.0f);
        }
    }
    __syncthreads();

    // ---- GEMM2: [16,128] x [128,128] ----
#pragma unroll
    for (int t = 0; t < 8; ++t) acc[t] = zero;

    for (int kk = 0; kk < 32; ++kk) {
        const int k0 = kk * 4 + 2 * half;
        v2f a;
        a.x = hb[l * HID + k0];
        a.y = hb[l * HID + k0 + 1];
        const float* wr = W2 + (size_t)k0 * HID;
#pragma unroll
        for (int t = 0; t < 8; ++t) {
            v2f b;
            b.x = wr[t * 16 + l];
            b.y = wr[HID + t * 16 + l];
            acc[t] = __builtin_amdgcn_wmma_f32_16x16x4_f32(
                false, a, false, b, (short)0, acc[t], false, false);
        }
    }

    // ---- bias + store ----
#pragma unroll
    for (int t = 0; t < 8; ++t) {
        const int col = t * 16 + l;
        const float bias = b2[col];
#pragma unroll
        for (int i = 0; i < 8; ++i) {
            out[(size_t)(rowBase + i + 8 * half) * HID + col] = acc[t][i] + bias;
        }
    }
}

// ---------------------------------------------------------------------------
extern "C" void kernel_launch(void* const* d_in, const int* in_sizes, int n_in,
                              void* d_out, int out_size, void* d_ws, size_t ws_size,
                              hipStream_t stream) {
    const float* x    = (const float*)d_in[0];
    const float* ea   = (const float*)d_in[1];
    const float* u    = (const float*)d_in[2];
    const float* W1   = (const float*)d_in[3];
    const float* b1   = (const float*)d_in[4];
    const float* W2   = (const float*)d_in[5];
    const float* b2   = (const float*)d_in[6];
    const int*   ei   = (const int*)d_in[7];
    const int*   bat  = (const int*)d_in[8];

    const int N  = in_sizes[8];
    const int E  = in_sizes[7] / 2;
    const int Bg = in_sizes[2] / HID;

    float* ws   = (float*)d_ws;
    float* nsum = ws;                                // [Bg,128]
    float* esum = ws + (size_t)Bg * HID;             // [Bg,128]
    float* ncnt = ws + (size_t)2 * Bg * HID;         // [Bg]
    float* ecnt = ncnt + Bg;                         // [Bg]

    // Zero accumulators (must happen every call: graph replays reuse ws).
    hipMemsetAsync(d_ws, 0, ((size_t)2 * Bg * HID + 2 * (size_t)Bg) * sizeof(float),
                   stream);

    // Node scatter (sorted batch -> register run-length accumulation).
    const int chunk = 256;
    const int ngrid = (N + chunk - 1) / chunk;
    node_scatter_kernel<<<ngrid, 128, 0, stream>>>(x, bat, nsum, ncnt, N, chunk);

    // Edge scatter (random destinations -> atomics).
    edge_scatter_kernel<<<4096, 128, 0, stream>>>(ea, ei, bat, esum, ecnt, E);

    // Means.
    const long total = (long)Bg * HID;
    finalize_mean_kernel<<<(int)((total + 255) / 256), 256, 0, stream>>>(
        nsum, esum, ncnt, ecnt, total);

    // Fused MLP (WMMA fp32).
    const int mgrid = (Bg + 63) / 64;
    mlp_wmma_kernel<<<mgrid, 128, 0, stream>>>(u, nsum, esum, W1, b1, W2, b2,
                                               (float*)d_out, Bg);
}